// GAT_56118042689980
// MI455X (gfx1250) — compile-verified
//
#include <hip/hip_runtime.h>
#include <hip/hip_bf16.h>
#include <math.h>

typedef __attribute__((ext_vector_type(2))) float v2f;
typedef __attribute__((ext_vector_type(8))) float v8f;

#define NEG_SLOPE 0.2f

// ---------------------------------------------------------------------------
// utility kernels
// ---------------------------------------------------------------------------
__global__ void fill_kernel(float* __restrict__ p, float v, size_t n) {
  size_t i = (size_t)blockIdx.x * blockDim.x + threadIdx.x;
  if (i < n) p[i] = v;
}

__global__ void add_bias_kernel(float* __restrict__ x, const float* __restrict__ bias,
                                size_t total, int F) {
  size_t i = (size_t)blockIdx.x * blockDim.x + threadIdx.x;
  if (i < total) x[i] += bias[i % F];
}

// float atomic max built from integer atomics (sound for the full float range):
// positive values: signed int max; negative values: unsigned int min.
__device__ __forceinline__ void atomicMaxF(float* addr, float val) {
  if (val >= 0.f) {
    atomicMax((int*)addr, __float_as_int(val));
  } else {
    atomicMin((unsigned int*)addr, (unsigned int)__float_as_int(val));
  }
}

// ---------------------------------------------------------------------------
// fp32 WMMA GEMM:  Y[N x M] = (RELU_IN ? relu(X) : X)[N x K] @ W[K x M]
//
// * W (K x M, <= 64KB) is staged once per block into LDS with the gfx1250
//   async copy path: global_load_async_to_lds_b128 + s_wait_asynccnt.
//   LDS row stride = M+16 floats -> row starts step 16 banks, so the two
//   half-wave B reads (rows k..k+1 vs k+2..k+3) hit disjoint banks.
// * Each wave owns one 16-row strip and all MT = M/16 column tiles:
//   per k-step one 8B global A load feeds MT v_wmma_f32_16x16x4_f32 ops
//   whose B fragments come from LDS (ds_load).
// * N must be a multiple of 16 (100000 = 6250*16).
// ---------------------------------------------------------------------------
template<bool RELU_IN, int MT>
__global__ void gemm_wmma_f32(const float* __restrict__ X, const float* __restrict__ W,
                              float* __restrict__ Y, int Nrows, int K) {
  constexpr int M = MT * 16;
  constexpr int S = M + 16;                 // padded LDS row stride (floats)
  __shared__ float sW[128 * S];             // K <= 128

  // ---- stage W into LDS via async copy (all 256 threads, coalesced b128) ----
  {
    const int chunksPerRow = M / 4;         // 4 floats per b128 chunk
    const int totalChunks  = K * chunksPerRow;
    for (int idx = threadIdx.x; idx < totalChunks; idx += 256) {
      int row = idx / chunksPerRow;
      int cc  = idx % chunksPerRow;
      unsigned gofs = (unsigned)((row * M + cc * 4) * sizeof(float));
      unsigned lofs = (unsigned)(uintptr_t)(&sW[row * S + cc * 4]);
      asm volatile("global_load_async_to_lds_b128 %0, %1, %2"
                   :: "v"(lofs), "v"(gofs), "s"(W) : "memory");
    }
    asm volatile("s_wait_asynccnt 0x0" ::: "memory");
  }
  __syncthreads();

  const int wave = threadIdx.x >> 5;
  const int lane = threadIdx.x & 31;
  const int rowTile = blockIdx.x * 8 + wave;
  const int numRowTiles = Nrows >> 4;
  if (rowTile >= numRowTiles) return;       // wave-uniform; after the barrier
  const int rowBase = rowTile << 4;
  const int half = lane >> 4;               // 0: k0/k1 + C rows 0-7 ; 1: k2/k3 + C rows 8-15
  const int l15  = lane & 15;
  const int k0   = half << 1;

  v8f acc[MT];
  v8f zero = {};
#pragma unroll
  for (int mt = 0; mt < MT; ++mt) acc[mt] = zero;

  const float* Xrow = X + (size_t)(rowBase + l15) * K + k0;   // A: row l15, k offset k0

  for (int k = 0; k < K; k += 4) {
    float ax = Xrow[k];
    float ay = Xrow[k + 1];
    if (RELU_IN) { ax = fmaxf(ax, 0.f); ay = fmaxf(ay, 0.f); }
    v2f a; a.x = ax; a.y = ay;
    const float* w0 = &sW[(k + k0)     * S + l15];
    const float* w1 = &sW[(k + k0 + 1) * S + l15];
#pragma unroll
    for (int mt = 0; mt < MT; ++mt) {
      v2f b; b.x = w0[mt * 16]; b.y = w1[mt * 16];
      acc[mt] = __builtin_amdgcn_wmma_f32_16x16x4_f32(
          /*neg_a=*/false, a, /*neg_b=*/false, b,
          /*c_mod=*/(short)0, acc[mt], /*reuse_a=*/false, /*reuse_b=*/false);
    }
  }

  // C/D layout: VGPR i -> row i (lanes 0-15) / row i+8 (lanes 16-31), col = lane&15
#pragma unroll
  for (int mt = 0; mt < MT; ++mt) {
#pragma unroll
    for (int i = 0; i < 8; ++i) {
      int row = rowBase + i + (half << 3);
      Y[(size_t)row * M + mt * 16 + l15] = acc[mt][i];
    }
  }
}

// ---------------------------------------------------------------------------
// per-node attention scores, wave per (n,h): coalesced loads + shuffle reduce
// a_src[n,h] = dot(xw[n,h,:], att_src[h,:]) ; same for a_dst.
// ---------------------------------------------------------------------------
__global__ void attn_scores_kernel(const float* __restrict__ xw,
                                   const float* __restrict__ att_src,
                                   const float* __restrict__ att_dst,
                                   float* __restrict__ a_src, float* __restrict__ a_dst,
                                   int Nn, int Hh) {
  int w    = blockIdx.x * (blockDim.x >> 5) + (threadIdx.x >> 5);  // w = n*Hh + h
  int lane = threadIdx.x & 31;
  if (w >= Nn * Hh) return;
  int h = w % Hh;
  const float* xp = xw + (size_t)w * 64;
  const float* as = att_src + h * 64;
  const float* ad = att_dst + h * 64;
  float v0 = xp[lane], v1 = xp[lane + 32];
  float s = fmaf(v0, as[lane], v1 * as[lane + 32]);
  float d = fmaf(v0, ad[lane], v1 * ad[lane + 32]);
  for (int off = 16; off; off >>= 1) {
    s += __shfl_down(s, off);
    d += __shfl_down(d, off);
  }
  if (lane == 0) { a_src[w] = s; a_dst[w] = d; }
}

// ---------------------------------------------------------------------------
// edge pass 1: e = leaky_relu(a_src[src]+a_dst[dst]); segment max via atomics
// ---------------------------------------------------------------------------
__global__ void edge_logits_kernel(const int* __restrict__ ei, int E, int Next,
                                   const float* __restrict__ a_src,
                                   const float* __restrict__ a_dst,
                                   float* __restrict__ ebuf, float* __restrict__ mmax, int Hh) {
  int e = blockIdx.x * blockDim.x + threadIdx.x;
  if (e >= Next) return;
  int s = (e < E) ? ei[e]     : (e - E);
  int d = (e < E) ? ei[E + e] : (e - E);
  for (int h = 0; h < Hh; ++h) {
    float v = a_src[s * Hh + h] + a_dst[d * Hh + h];
    v = (v > 0.f) ? v : NEG_SLOPE * v;
    ebuf[(size_t)e * Hh + h] = v;
    atomicMaxF(&mmax[d * Hh + h], v);
  }
}

// edge pass 2: p = exp(e - m[dst]); denom[dst] += p  (p stored in-place)
__global__ void edge_expsum_kernel(const int* __restrict__ ei, int E, int Next,
                                   float* __restrict__ ebuf,
                                   const float* __restrict__ mmax,
                                   float* __restrict__ denom, int Hh) {
  int e = blockIdx.x * blockDim.x + threadIdx.x;
  if (e >= Next) return;
  int d = (e < E) ? ei[E + e] : (e - E);
  for (int h = 0; h < Hh; ++h) {
    float m = mmax[d * Hh + h];
    if (!__builtin_isfinite(m)) m = 0.f;
    float p = __expf(ebuf[(size_t)e * Hh + h] - m);
    ebuf[(size_t)e * Hh + h] = p;
    atomicAdd(&denom[d * Hh + h], p);
  }
}

// edge pass 3: alpha = p/denom (stored into alpha region in-place);
// wave per edge: lanes stride the F = Hh*D features; coalesced atomicAdds.
__global__ void edge_aggregate_kernel(const int* __restrict__ ei, int E, int Next,
                                      float* __restrict__ ebuf /* p in, alpha out */,
                                      const float* __restrict__ denom,
                                      const float* __restrict__ xw,
                                      float* __restrict__ xout,
                                      int Hh, int F) {
  int w    = (int)(blockIdx.x << 3) + (threadIdx.x >> 5);
  int lane = threadIdx.x & 31;
  if (w >= Next) return;
  int e = w;
  int s = (e < E) ? ei[e]     : (e - E);
  int d = (e < E) ? ei[E + e] : (e - E);
  float alpha[2];
  for (int h = 0; h < Hh; ++h)
    alpha[h] = ebuf[(size_t)e * Hh + h] / (denom[d * Hh + h] + 1e-16f);
  if (lane < Hh) ebuf[(size_t)e * Hh + lane] = alpha[lane];  // final alpha output
  const int Dd = F / Hh;
  const float* src = xw + (size_t)s * F;
  float* dst = xout + (size_t)d * F;
  for (int f = lane; f < F; f += 32)
    atomicAdd(&dst[f], alpha[f / Dd] * src[f]);
}

// ---------------------------------------------------------------------------
// final = (x0 + mean_h(x1) + mean_h(x2) + x3) / 4
// ---------------------------------------------------------------------------
__global__ void finalize_kernel(const float* __restrict__ x0, const float* __restrict__ x1,
                                const float* __restrict__ x2, const float* __restrict__ x3,
                                float* __restrict__ out_final, int Nn) {
  int idx = blockIdx.x * blockDim.x + threadIdx.x;   // over Nn*64
  if (idx >= Nn * 64) return;
  int n  = idx >> 6;
  int dd = idx & 63;
  size_t b = (size_t)n << 7;   // row base in the N x 128 buffers
  float e1 = 0.5f * (x1[b + dd] + x1[b + 64 + dd]);
  float e2 = 0.5f * (x2[b + dd] + x2[b + 64 + dd]);
  out_final[idx] = (x0[idx] + e1 + e2 + x3[idx]) * 0.25f;
}

// ---------------------------------------------------------------------------
// host-side layer driver
// ---------------------------------------------------------------------------
static void run_gat_layer(const float* Xin, bool reluIn, int K, int M,
                          const float* W, const float* att_s, const float* att_d,
                          const float* bias,
                          float* xw, float* asrc, float* adst, float* mmax, float* denom,
                          float* xlayer, float* alpha, const int* ei,
                          int Nn, int Hh, int E, int Next, hipStream_t stream) {
  const int gemmBlocks = (Nn / 16 + 7) / 8;
  if (M == 128) {
    if (reluIn)
      gemm_wmma_f32<true , 8><<<gemmBlocks, 256, 0, stream>>>(Xin, W, xw, Nn, K);
    else
      gemm_wmma_f32<false, 8><<<gemmBlocks, 256, 0, stream>>>(Xin, W, xw, Nn, K);
  } else { // M == 64
    if (reluIn)
      gemm_wmma_f32<true , 4><<<gemmBlocks, 256, 0, stream>>>(Xin, W, xw, Nn, K);
    else
      gemm_wmma_f32<false, 4><<<gemmBlocks, 256, 0, stream>>>(Xin, W, xw, Nn, K);
  }

  int nh = Nn * Hh;
  attn_scores_kernel<<<(nh + 7) / 8, 256, 0, stream>>>(xw, att_s, att_d, asrc, adst, Nn, Hh);
  fill_kernel<<<(nh + 255) / 256, 256, 0, stream>>>(mmax, -INFINITY, (size_t)nh);
  fill_kernel<<<(nh + 255) / 256, 256, 0, stream>>>(denom, 0.f, (size_t)nh);
  size_t tot = (size_t)Nn * M;
  fill_kernel<<<(int)((tot + 255) / 256), 256, 0, stream>>>(xlayer, 0.f, tot);

  edge_logits_kernel<<<(Next + 255) / 256, 256, 0, stream>>>(ei, E, Next, asrc, adst, alpha, mmax, Hh);
  edge_expsum_kernel<<<(Next + 255) / 256, 256, 0, stream>>>(ei, E, Next, alpha, mmax, denom, Hh);
  edge_aggregate_kernel<<<(Next + 7) / 8, 256, 0, stream>>>(ei, E, Next, alpha, denom, xw, xlayer, Hh, M);
  add_bias_kernel<<<(int)((tot + 255) / 256), 256, 0, stream>>>(xlayer, bias, tot, M);
}

extern "C" void kernel_launch(void* const* d_in, const int* in_sizes, int n_in,
                              void* d_out, int out_size, void* d_ws, size_t ws_size,
                              hipStream_t stream) {
  constexpr int I_ = 40000, U_ = 30000, Fh_ = 30000;
  constexpr int Nn = I_ + U_ + Fh_;          // 100000
  constexpr int Dd = 64, Hh = 2;
  constexpr int Eg = 1200000, Next = Eg + Nn; // 1.3M with self loops
  constexpr size_t ND  = (size_t)Nn * Dd;     // 6.4e6
  constexpr size_t NHD = (size_t)Nn * Hh * Dd;// 12.8e6
  constexpr size_t NH  = (size_t)Nn * Hh;

  const int*   ei       = (const int*)d_in[0];
  const float* emb_item = (const float*)d_in[1];
  const float* emb_user = (const float*)d_in[2];
  const float* emb_fet  = (const float*)d_in[3];
  const float* W1  = (const float*)d_in[4];
  const float* as1 = (const float*)d_in[5];
  const float* ad1 = (const float*)d_in[6];
  const float* b1  = (const float*)d_in[7];
  const float* W2  = (const float*)d_in[8];
  const float* as2 = (const float*)d_in[9];
  const float* ad2 = (const float*)d_in[10];
  const float* b2  = (const float*)d_in[11];
  const float* W3  = (const float*)d_in[12];
  const float* as3 = (const float*)d_in[13];
  const float* ad3 = (const float*)d_in[14];
  const float* b3  = (const float*)d_in[15];

  // workspace carve (floats): ~208 MB
  float* ws    = (float*)d_ws;
  float* x0    = ws;             // ND
  float* xw    = x0 + ND;        // NHD (per-layer xw)
  float* x1    = xw + NHD;       // NHD
  float* x2    = x1 + NHD;       // NHD
  float* x3    = x2 + NHD;       // ND
  float* asrc  = x3 + ND;        // NH
  float* adst  = asrc + NH;      // NH
  float* mmax  = adst + NH;      // NH
  float* denom = mmax + NH;      // NH

  // output carve (reference return order)
  float* out      = (float*)d_out;
  float* o_item   = out;                               // I*D
  float* o_user   = o_item + (size_t)I_ * Dd;          // U*D
  float* o_fet    = o_user + (size_t)U_ * Dd;          // Fh*D
  float* o_final  = o_fet  + (size_t)Fh_ * Dd;         // N*D
  float* o_alpha1 = o_final + ND;                      // Next*2
  float* o_alpha2 = o_alpha1 + (size_t)Next * Hh;      // Next*2
  float* o_alpha3 = o_alpha2 + (size_t)Next * Hh;      // Next*1

  // x0 = concat(item, user, fet)
  hipMemcpyAsync(x0,                      emb_item, (size_t)I_  * Dd * sizeof(float), hipMemcpyDeviceToDevice, stream);
  hipMemcpyAsync(x0 + (size_t)I_ * Dd,    emb_user, (size_t)U_  * Dd * sizeof(float), hipMemcpyDeviceToDevice, stream);
  hipMemcpyAsync(x0 + (size_t)(I_+U_)*Dd, emb_fet,  (size_t)Fh_ * Dd * sizeof(float), hipMemcpyDeviceToDevice, stream);

  // layer 1: x0 (N x 64) -> x1 (N x 128), heads=2
  run_gat_layer(x0, /*relu=*/false, /*K=*/64, /*M=*/128,
                W1, as1, ad1, b1, xw, asrc, adst, mmax, denom,
                x1, o_alpha1, ei, Nn, 2, Eg, Next, stream);

  // layer 2: relu(x1) (N x 128) -> x2 (N x 128), heads=2
  run_gat_layer(x1, /*relu=*/true, /*K=*/128, /*M=*/128,
                W2, as2, ad2, b2, xw, asrc, adst, mmax, denom,
                x2, o_alpha2, ei, Nn, 2, Eg, Next, stream);

  // layer 3: relu(x2) (N x 128) -> x3 (N x 64), heads=1 (mean == identity)
  run_gat_layer(x2, /*relu=*/true, /*K=*/128, /*M=*/64,
                W3, as3, ad3, b3, xw, asrc, adst, mmax, denom,
                x3, o_alpha3, ei, Nn, 1, Eg, Next, stream);

  // final = (x0 + mean(x1) + mean(x2) + x3) / 4
  finalize_kernel<<<(int)((ND + 255) / 256), 256, 0, stream>>>(x0, x1, x2, x3, o_final, Nn);

  // splits
  hipMemcpyAsync(o_item, o_final,                        (size_t)I_  * Dd * sizeof(float), hipMemcpyDeviceToDevice, stream);
  hipMemcpyAsync(o_user, o_final + (size_t)I_ * Dd,      (size_t)U_  * Dd * sizeof(float), hipMemcpyDeviceToDevice, stream);
  hipMemcpyAsync(o_fet,  o_final + (size_t)(I_+U_) * Dd, (size_t)Fh_ * Dd * sizeof(float), hipMemcpyDeviceToDevice, stream);
}